// EncodingP_4836133175527
// MI455X (gfx1250) — compile-verified
//
#include <hip/hip_runtime.h>
#include <stdint.h>

// Problem constants (match the reference)
#define BN 8
#define DD 256
#define NN 1024
#define KK 32

typedef float v2f __attribute__((ext_vector_type(2)));
typedef float v8f __attribute__((ext_vector_type(8)));

__device__ __forceinline__ float shflx(float v, int mask) {
    return __shfl_xor(v, mask, 32);   // wave32 on gfx1250
}

// --- CDNA5 async global->LDS copies (ASYNCcnt-tracked, per-lane scatter) ----
__device__ __forceinline__ uint32_t lds_off(const void* p) {
    // Generic pointers to LDS carry the shared aperture in the high 32 bits;
    // the low 32 bits are the wave-relative LDS byte offset (ISA 10.2).
    return (uint32_t)(uintptr_t)p;
}
__device__ __forceinline__ void async_g2l_b32(uint32_t lds, const void* g) {
    asm volatile("global_load_async_to_lds_b32 %0, %1, off"
                 :: "v"(lds), "v"(g) : "memory");
}
__device__ __forceinline__ void async_g2l_b128(uint32_t lds, const void* g) {
    asm volatile("global_load_async_to_lds_b128 %0, %1, off"
                 :: "v"(lds), "v"(g) : "memory");
}

// ---------------------------------------------------------------------------
// Kernel 0: zero the S accumulator region of the workspace (B*K floats)
// ---------------------------------------------------------------------------
__global__ void enc_zero(float* __restrict__ S) {
    S[threadIdx.x] = 0.0f;            // launched with exactly BN*KK threads
}

// ---------------------------------------------------------------------------
// Kernel 1: per (b, n) compute softmax_k( scale[k]*||x_n - c_k||^2 )
//   dot[k][n] = sum_d C[k,d] * X[b,d,n]  via V_WMMA_F32_16X16X4_F32
//   A stored to ws in (B, K, N) layout; S[b,k] accumulated via atomics.
// ---------------------------------------------------------------------------
__global__ __launch_bounds__(256) void enc_assign(
    const float* __restrict__ Xg,      // (B, D, N)
    const float* __restrict__ Cg,      // (K, D)
    const float* __restrict__ scaleg,  // (K)
    float* __restrict__ Aws,           // (B, K, N)
    float* __restrict__ Sws) {         // (B, K)
    __shared__ float Clds[KK * DD];    // 32 KB codewords
    __shared__ float cnorm[KK];
    __shared__ float slds[KK];

    const int t = threadIdx.x;
    const int b = blockIdx.x >> 3;
    const int nchunk = blockIdx.x & 7;

    // Stage codewords into LDS (coalesced b128)
    for (int i = t * 4; i < KK * DD; i += 256 * 4)
        *(float4*)&Clds[i] = *(const float4*)&Cg[i];
    if (t < KK) { cnorm[t] = 0.0f; slds[t] = scaleg[t]; }
    __syncthreads();

    // ||c_k||^2 via LDS float atomics (ds_add_f32)
    {
        const int k = t & 31, seg = t >> 5;     // 8 segments of 32 d's
        float p = 0.0f;
        #pragma unroll
        for (int d = 0; d < 32; ++d) {
            float c = Clds[k * DD + seg * 32 + d];
            p += c * c;
        }
        atomicAdd(&cnorm[k], p);
    }
    __syncthreads();

    const int wave = t >> 5;
    const int lane = t & 31;
    const int half = lane >> 4;       // 0: lanes 0-15, 1: lanes 16-31
    const int l15  = lane & 15;
    const int n0   = nchunk * 128 + wave * 16;
    const int n    = n0 + l15;        // this lane's column (both halves)

    const float* Xb = Xg + (size_t)b * DD * NN;

    v8f acc0 = {};                    // dot tile, k = 0..15
    v8f acc1 = {};                    // dot tile, k = 16..31
    float xx = 0.0f;                  // partial ||x_n||^2 (half the d's)

    for (int d0 = 0; d0 < DD; d0 += 4) {
        const int ds = d0 + half * 2; // K-cols {0,1} vs {2,3} per ISA layout
        v2f bfrag;
        const float bx = Xb[(size_t)ds * NN + n];
        const float by = Xb[(size_t)(ds + 1) * NN + n];
        bfrag.x = bx; bfrag.y = by;
        xx += bx * bx + by * by;

        v2f a0 = *(const v2f*)&Clds[l15 * DD + ds];
        v2f a1 = *(const v2f*)&Clds[(16 + l15) * DD + ds];

        acc0 = __builtin_amdgcn_wmma_f32_16x16x4_f32(false, a0, false, bfrag,
                                                     (short)0, acc0, false, false);
        acc1 = __builtin_amdgcn_wmma_f32_16x16x4_f32(false, a1, false, bfrag,
                                                     (short)0, acc1, false, false);
    }

    const float xxf = xx + shflx(xx, 16);   // full ||x_n||^2

    // Per-lane: 16 of the 32 k-values of column n (other 16 in lane^16)
    float sl[16], av[16];
    #pragma unroll
    for (int j = 0; j < 8; ++j) {
        const int k0 = j + half * 8;
        const int k1 = 16 + j + half * 8;
        sl[j]     = slds[k0] * (xxf - 2.0f * acc0[j] + cnorm[k0]);
        sl[8 + j] = slds[k1] * (xxf - 2.0f * acc1[j] + cnorm[k1]);
    }
    float m = sl[0];
    #pragma unroll
    for (int j = 1; j < 16; ++j) m = fmaxf(m, sl[j]);
    m = fmaxf(m, shflx(m, 16));
    float s = 0.0f;
    #pragma unroll
    for (int j = 0; j < 16; ++j) { av[j] = __expf(sl[j] - m); s += av[j]; }
    const float inv = 1.0f / (s + shflx(s, 16));

    // Store A in (B,K,N) layout (coalesced 64B rows) + accumulate S[b,k]
    float* Ab = Aws + (size_t)b * KK * NN;
    #pragma unroll
    for (int j = 0; j < 8; ++j) {
        const int k0 = j + half * 8;
        const int k1 = 16 + j + half * 8;
        const float a0v = av[j] * inv;
        const float a1v = av[8 + j] * inv;
        Ab[(size_t)k0 * NN + n] = a0v;
        Ab[(size_t)k1 * NN + n] = a1v;
        float r0 = a0v, r1 = a1v;
        r0 += shflx(r0, 1); r0 += shflx(r0, 2); r0 += shflx(r0, 4); r0 += shflx(r0, 8);
        r1 += shflx(r1, 1); r1 += shflx(r1, 2); r1 += shflx(r1, 4); r1 += shflx(r1, 8);
        if (l15 == 0) {
            atomicAdd(&Sws[b * KK + k0], r0);
            atomicAdd(&Sws[b * KK + k1], r1);
        }
    }
}

// ---------------------------------------------------------------------------
// Kernel 2: E[b,k,d] = sum_n A[b,n,k]*X[b,d,n] - S[b,k]*C[k,d]
// Double-buffered async global->LDS staging (ASYNCcnt), WMMA reduction.
// Block: 256 threads (8 waves): wave = (kt in 0..1) x (dt in 0..3).
// Grid:  (D/64, B). Each wave: 16(k) x 16(d) tile, reduce N=1024 in 16 tiles.
// ---------------------------------------------------------------------------
#define NT 64                       // n super-tile
__global__ __launch_bounds__(256) void enc_aggregate(
    const float* __restrict__ Xg,      // (B, D, N)
    const float* __restrict__ Cg,      // (K, D)
    const float* __restrict__ Aws,     // (B, K, N)
    const float* __restrict__ Sws,     // (B, K)
    float* __restrict__ Eout) {        // (B, K, D)
    __shared__ float As2[2][KK * NT];  // 2 x 8 KB : A^T tile, (k, n)
    __shared__ float Xs2[2][NT * 64];  // 2 x 16 KB: X^T tile, (n, d)

    const int t = threadIdx.x;
    const int b = blockIdx.y;
    const int dbase = blockIdx.x * 64;
    const int wave = t >> 5, lane = t & 31;
    const int half = lane >> 4, l15 = lane & 15;
    const int kt = wave >> 2;          // k tile: 0 or 1
    const int dd0 = (wave & 3) * 16;   // d tile within chunk

    const float* Xb = Xg + (size_t)b * DD * NN;
    const float* Ab = Aws + (size_t)b * KK * NN;

    // Per-thread staging geometry (identical every super-tile)
    const int ka0 = t >> 4,         ca0 = (t & 15) * 4;          // A float4 #t
    const int ka1 = (t + 256) >> 4, ca1 = ((t + 256) & 15) * 4;  // A float4 #t+256
    const int dx  = t & 63;          // this thread's d column for X transpose
    const int ngx = (t >> 6) * 16;   // 16 n-rows handled by this thread

    // Issue all 18 async copies (2 x b128 for A, 16 x b32 transposing X)
    auto stage = [&](int buf, int ns) {
        uint32_t asb = lds_off(&As2[buf][0]);
        uint32_t xsb = lds_off(&Xs2[buf][0]);
        async_g2l_b128(asb + (uint32_t)(ka0 * NT + ca0) * 4u,
                       &Ab[(size_t)ka0 * NN + ns + ca0]);
        async_g2l_b128(asb + (uint32_t)(ka1 * NT + ca1) * 4u,
                       &Ab[(size_t)ka1 * NN + ns + ca1]);
        const float* src = &Xb[(size_t)(dbase + dx) * NN + ns + ngx];
        #pragma unroll
        for (int i = 0; i < 16; ++i)
            async_g2l_b32(xsb + (uint32_t)((ngx + i) * 64 + dx) * 4u, src + i);
    };

    v8f acc = {};
    stage(0, 0);

    for (int it = 0; it < NN / NT; ++it) {
        const int buf = it & 1;
        if (it + 1 < NN / NT) {
            stage(buf ^ 1, (it + 1) * NT);
            // prior tile's 18 ops precede the 18 just issued; in-order completion
            asm volatile("s_wait_asynccnt 0x12" ::: "memory");
        } else {
            asm volatile("s_wait_asynccnt 0x0" ::: "memory");
        }
        __syncthreads();

        const float* As = &As2[buf][0];
        const float* Xs = &Xs2[buf][0];
        #pragma unroll
        for (int nn0 = 0; nn0 < NT; nn0 += 4) {
            const int nsel = nn0 + half * 2;          // ISA A/B half-lane K split
            v2f a = *(const v2f*)&As[(kt * 16 + l15) * NT + nsel];
            v2f bf;
            bf.x = Xs[nsel * 64 + dd0 + l15];
            bf.y = Xs[(nsel + 1) * 64 + dd0 + l15];
            acc = __builtin_amdgcn_wmma_f32_16x16x4_f32(false, a, false, bf,
                                                        (short)0, acc, false, false);
        }
        __syncthreads();   // all waves done with buf before it is restaged
    }

    // Epilogue: E = acc - S[b,k]*C[k,d]; disjoint coalesced stores
    #pragma unroll
    for (int j = 0; j < 8; ++j) {
        const int k = kt * 16 + j + half * 8;         // C/D row layout: j / j+8
        const int d = dbase + dd0 + l15;
        const float sv = Sws[b * KK + k];
        const float cv = Cg[(size_t)k * DD + d];
        Eout[((size_t)b * KK + k) * DD + d] = acc[j] - sv * cv;
    }
}

// ---------------------------------------------------------------------------
extern "C" void kernel_launch(void* const* d_in, const int* in_sizes, int n_in,
                              void* d_out, int out_size, void* d_ws, size_t ws_size,
                              hipStream_t stream) {
    const float* X     = (const float*)d_in[0];   // (B, D, H, W) fp32
    const float* C     = (const float*)d_in[1];   // (K, D) fp32
    const float* scale = (const float*)d_in[2];   // (K) fp32
    float* out = (float*)d_out;                   // (B, K, D) fp32

    float* Sws = (float*)d_ws;                    // B*K floats
    float* Aws = Sws + BN * KK;                   // B*K*N floats (~1 MB)

    enc_zero<<<1, BN * KK, 0, stream>>>(Sws);
    enc_assign<<<BN * (NN / 128), 256, 0, stream>>>(X, C, scale, Aws, Sws);
    enc_aggregate<<<dim3(DD / 64, BN), 256, 0, stream>>>(X, C, Aws, Sws, out);
}